// DenseCapsule_52544629899906
// MI455X (gfx1250) — compile-verified
//
#include <hip/hip_runtime.h>
#include <hip/hip_bf16.h>
#include <math.h>

// Dims
#define Bdim 32
#define NIN  2048
#define DIN  16
#define Jcap 64
#define Ccap 32
#define BJ   (Bdim * Jcap)            // 2048
#define BC   (Bdim * Ccap)            // 1024: stride per i in uh[j,i,b,c]
#define EPSV 1e-7f

typedef __attribute__((ext_vector_type(2))) float v2f;
typedef __attribute__((ext_vector_type(4))) float v4f;
typedef __attribute__((ext_vector_type(8))) float v8f;

// ---------------------------------------------------------------------------
// Kernel 1: u_hat[j,i,b,c] = sum_d u[b,i,d] * W[i,j,c,d]
// One block per input capsule i. 8 waves; each wave owns 16 of the 128 N-tiles
// (N = j*32+c), computes two 16x16 M-tiles (batch 0-15 / 16-31) via
// 4x V_WMMA_F32_16X16X4_F32 (K=16). Output tile for fixed (j,i) is a
// contiguous 4KB window -> immediate-offset stores, no 64-bit addr chains.
// ---------------------------------------------------------------------------
__global__ __launch_bounds__(256) void caps_gemm(const float* __restrict__ x,
                                                 const float* __restrict__ W,
                                                 float* __restrict__ uh) {
    const int i      = blockIdx.x;
    const int wave   = threadIdx.x >> 5;
    const int lane   = threadIdx.x & 31;
    const int lanelo = lane & 15;
    const int lanehi = lane >> 4;           // 0 or 1

    // Preload A fragments: 16x4 f32 tile layout:
    //   lanes 0-15: M = lanelo; VGPR0 = K=kb, VGPR1 = K=kb+1, kb = ks*4 + lanehi*2
    v2f a[2][4];
#pragma unroll
    for (int mt = 0; mt < 2; ++mt) {
        const int m = mt * 16 + lanelo;
        const float* up = x + ((size_t)m * NIN + i) * DIN;   // x[b,i,d,0]
#pragma unroll
        for (int ks = 0; ks < 4; ++ks) {
            const int kb = ks * 4 + lanehi * 2;
            a[mt][ks].x = up[kb];
            a[mt][ks].y = up[kb + 1];
        }
    }

    for (int nt = wave; nt < 128; nt += 8) {
        const int j     = nt >> 1;            // n = nt*16 + lanelo; j = n>>5
        const int chalf = (nt & 1) * 16;      // c = chalf + lanelo
        const int c     = chalf + lanelo;
        const float* wp = W + (((size_t)i * Jcap + j) * Ccap + c) * DIN;

        // Prefetch next W tile for this wave (streams W once -> NT path)
        if (nt + 8 < 128) {
            const int jn = (nt + 8) >> 1;
            const int cn = ((nt + 8) & 1) * 16 + lanelo;
            __builtin_prefetch(W + (((size_t)i * Jcap + jn) * Ccap + cn) * DIN, 0, 1);
        }

        v8f acc0 = {};
        v8f acc1 = {};
#pragma unroll
        for (int ks = 0; ks < 4; ++ks) {
            const int kb = ks * 4 + lanehi * 2;
            const v2f bf = __builtin_nontemporal_load((const v2f*)(wp + kb));
            acc0 = __builtin_amdgcn_wmma_f32_16x16x4_f32(
                false, a[0][ks], false, bf, (short)0, acc0, false, false);
            acc1 = __builtin_amdgcn_wmma_f32_16x16x4_f32(
                false, a[1][ks], false, bf, (short)0, acc1, false, false);
        }

        // C/D layout: VGPR r holds M = r + lanehi*8, N = lanelo (within tile).
        // uh[j,i,b,c] -> lane base + r*32 floats; second M-tile at +512 floats.
        float* outp = uh + ((size_t)j * NIN + i) * BC
                         + (size_t)lanehi * 256 + chalf + lanelo;
#pragma unroll
        for (int r = 0; r < 8; ++r) {
            __builtin_nontemporal_store(acc0[r], outp + r * Ccap);
            __builtin_nontemporal_store(acc1[r], outp + r * Ccap + 512);
        }
    }
}

// ---------------------------------------------------------------------------
// Kernel 2: routing logits update + softmax statistics.
// One block per (b,j):  bmat[bj,i] (+)= dot_c(v[bj,:], uh[j,i,b,:]);
// then block max over i and sum of exp(b - max).
// ---------------------------------------------------------------------------
__global__ __launch_bounds__(256) void caps_update(const float* __restrict__ uh,
                                                   const float* __restrict__ v,
                                                   float* __restrict__ bmat,
                                                   float* __restrict__ stats,
                                                   int firstIter) {
    const int bj = blockIdx.x;
    const int t  = threadIdx.x;
    const int b_ = bj >> 6;
    const int j  = bj & 63;
    const float* base = uh + (size_t)j * NIN * BC + (size_t)b_ * Ccap;
    float* brow       = bmat + (size_t)bj * NIN;

    __shared__ float vsh[Ccap];
    if (t < Ccap) vsh[t] = v[(size_t)bj * Ccap + t];
    __syncthreads();

    float lmax = -3.402823466e+38f;
    for (int i = t; i < NIN; i += 256) {
        const float* p = base + (size_t)i * BC;
        float dot = 0.f;
#pragma unroll
        for (int c = 0; c < Ccap; c += 4) {
            const v4f u4 = __builtin_nontemporal_load((const v4f*)(p + c));
            dot += u4.x * vsh[c] + u4.y * vsh[c + 1] +
                   u4.z * vsh[c + 2] + u4.w * vsh[c + 3];
        }
        const float bn = (firstIter ? 0.f : brow[i]) + dot;
        brow[i] = bn;
        lmax = fmaxf(lmax, bn);
    }

    __shared__ float red[256];
    red[t] = lmax;
    __syncthreads();
    for (int s = 128; s > 0; s >>= 1) {
        if (t < s) red[t] = fmaxf(red[t], red[t + s]);
        __syncthreads();
    }
    const float m = red[0];
    __syncthreads();

    float lsum = 0.f;
    for (int i = t; i < NIN; i += 256) lsum += expf(brow[i] - m);
    red[t] = lsum;
    __syncthreads();
    for (int s = 128; s > 0; s >>= 1) {
        if (t < s) red[t] += red[t + s];
        __syncthreads();
    }
    if (t == 0) {
        stats[2 * bj]     = m;
        stats[2 * bj + 1] = red[0];
    }
}

// ---------------------------------------------------------------------------
// Kernel 3: weighted sum over i + squash.
// useB==0: uniform weights 1/2048 (iteration 0, softmax of zeros).
// ---------------------------------------------------------------------------
__global__ __launch_bounds__(256) void caps_reduce(const float* __restrict__ uh,
                                                   const float* __restrict__ bmat,
                                                   const float* __restrict__ stats,
                                                   float* __restrict__ vout,
                                                   int useB) {
    const int bj = blockIdx.x;
    const int t  = threadIdx.x;
    const int c  = t & 31;
    const int g  = t >> 5;                    // 8 wave-groups
    const int b_ = bj >> 6;
    const int j  = bj & 63;
    const float* base = uh + (size_t)j * NIN * BC + (size_t)b_ * Ccap;
    const float* brow = bmat + (size_t)bj * NIN;

    float m = 0.f, inv = 0.f;
    if (useB) {
        m   = stats[2 * bj];
        inv = 1.0f / stats[2 * bj + 1];
    }

    float acc = 0.f;
    for (int i = g; i < NIN; i += 8) {
        const float w = useB ? (expf(brow[i] - m) * inv) : (1.0f / (float)NIN);
        acc += w * __builtin_nontemporal_load(base + (size_t)i * BC + c);
    }

    __shared__ float red[256];
    __shared__ float sv[Ccap];
    __shared__ float nsh;
    red[t] = acc;
    __syncthreads();
    if (t < Ccap) {
        float s = 0.f;
        for (int gg = 0; gg < 8; ++gg) s += red[gg * 32 + t];
        sv[t] = s + EPSV;                      // squash: v += eps
    }
    __syncthreads();
    if (t == 0) {
        float n = 0.f;
        for (int cc = 0; cc < Ccap; ++cc) n += sv[cc] * sv[cc];
        nsh = n;
    }
    __syncthreads();
    if (t < Ccap) {
        const float n = nsh;
        vout[(size_t)bj * Ccap + t] = sv[t] * (n / (1.0f + n) / sqrtf(n));
    }
}

// ---------------------------------------------------------------------------
extern "C" void kernel_launch(void* const* d_in, const int* in_sizes, int n_in,
                              void* d_out, int out_size, void* d_ws, size_t ws_size,
                              hipStream_t stream) {
    (void)in_sizes; (void)n_in; (void)out_size; (void)ws_size;
    const float* x = (const float*)d_in[0];   // (B, N_IN, D_IN, 1)
    const float* W = (const float*)d_in[1];   // (N_IN, J, C, D_IN)
    float* out = (float*)d_out;               // (B, J, C, 1) flat = bj*32+c

    char* ws = (char*)d_ws;
    const size_t uhBytes = (size_t)Jcap * NIN * Bdim * Ccap * sizeof(float); // 512 MB
    const size_t bBytes  = (size_t)BJ * NIN * sizeof(float);                 // 16 MB
    float* uh    = (float*)ws;
    float* bmat  = (float*)(ws + uhBytes);
    float* stats = (float*)(ws + uhBytes + bBytes);
    float* v     = stats + 2 * BJ;

    // 1) u_hat via WMMA f32 16x16x4
    caps_gemm<<<NIN, 256, 0, stream>>>(x, W, uh);

    // 2) iteration 0: uniform softmax -> mean -> squash
    caps_reduce<<<BJ, 256, 0, stream>>>(uh, bmat, stats, v, 0);

    // 3) iterations 1..4: b update + softmax stats, then weighted sum + squash
    for (int it = 1; it < 5; ++it) {
        caps_update<<<BJ, 256, 0, stream>>>(uh, v, bmat, stats, (it == 1) ? 1 : 0);
        caps_reduce<<<BJ, 256, 0, stream>>>(uh, bmat, stats,
                                            (it == 4) ? out : v, 1);
    }
}